// LossingDotProduct_70781061038125
// MI455X (gfx1250) — compile-verified
//
#include <hip/hip_runtime.h>
#include <math.h>

// CDNA5 / gfx1250, wave32.
typedef __attribute__((ext_vector_type(2))) float v2f;
typedef __attribute__((ext_vector_type(8))) float v8f;

#define DIM   512
#define KNEG  64
#define LLAB  20
#define NEG_INF_F (-1.0e9f)

__device__ __forceinline__ float wave_max_f(float x) {
#pragma unroll
  for (int m = 16; m >= 1; m >>= 1) x = fmaxf(x, __shfl_xor(x, m, 32));
  return x;
}
__device__ __forceinline__ float wave_sum_f(float x) {
#pragma unroll
  for (int m = 16; m >= 1; m >>= 1) x += __shfl_xor(x, m, 32);
  return x;
}

// One wave (32 lanes) per batch row b. Computes per_ex[b] and acc flag[b].
__global__ __launch_bounds__(256) void sims_softmax_kernel(
    const float* __restrict__ inputs_embed,     // [B, D]
    const float* __restrict__ labels_embed,     // [B, D]
    const float* __restrict__ labels,           // [B, L]
    const float* __restrict__ all_labels_embed, // [N, D]
    const float* __restrict__ all_labels,       // [N, L]
    const int*   __restrict__ neg_ids_in,       // [B, K]
    const int*   __restrict__ neg_ids_lb,       // [B, K]
    float* __restrict__ per_ex_out,             // [B]
    float* __restrict__ flag_out,               // [B]
    int Bn)
{
  const int wave = threadIdx.x >> 5;
  const int lane = threadIdx.x & 31;
  const int b = blockIdx.x * (blockDim.x >> 5) + wave;
  if (b >= Bn) return;  // whole-wave uniform exit

  const int n = lane & 15;                 // column within a 16-neg tile / A row id
  const int koff = (lane < 16) ? 0 : 2;    // K split across lane halves (ISA layout)
  const bool arow = (n < 2);               // lanes holding the u/v rows of A

  const float* u = inputs_embed + (size_t)b * DIM;
  const float* v = labels_embed + (size_t)b * DIM;
  const int* ids_lb = neg_ids_lb + (size_t)b * KNEG;
  const int* ids_in = neg_ids_in + (size_t)b * KNEG;
  const float* lab = labels + (size_t)b * LLAB;

  // Per-lane negative row pointers for 8 tiles (t0..3 = le-negs, t4..7 = ie-negs)
  // plus the NEG_INF "bad negative" additive masks.
  const float* pB[8];
  float badadd[8];
#pragma unroll
  for (int t = 0; t < 4; ++t) {
    const int id = ids_lb[t * 16 + n];
    pB[t] = all_labels_embed + (size_t)id * DIM;
    const float* nl = all_labels + (size_t)id * LLAB;
    bool eq = true;
#pragma unroll
    for (int j = 0; j < LLAB; ++j) eq = eq && (nl[j] == lab[j]);
    badadd[t] = eq ? NEG_INF_F : 0.0f;
  }
#pragma unroll
  for (int t = 0; t < 4; ++t) {
    const int id = ids_in[t * 16 + n];
    pB[4 + t] = inputs_embed + (size_t)id * DIM;
    const float* nl = labels + (size_t)id * LLAB;
    bool eq = true;
#pragma unroll
    for (int j = 0; j < LLAB; ++j) eq = eq && (nl[j] == lab[j]);
    badadd[4 + t] = eq ? NEG_INF_F : 0.0f;
  }

  // sim_pos = u . v (full-precision VALU + wave reduce)
  float sp = 0.0f;
  for (int d = lane; d < DIM; d += 32) sp = fmaf(u[d], v[d], sp);
  sp = wave_sum_f(sp);

  // WMMA accumulators: 8 tiles of 16x16 f32. Rows 0/1 of A = u/v; rows 2..15 zero.
  v8f c[8];
#pragma unroll
  for (int t = 0; t < 8; ++t)
#pragma unroll
    for (int e = 0; e < 8; ++e) c[t][e] = 0.0f;

  // All lanes load a valid address (u for the dead rows) and mask the value:
  // no EXEC manipulation in the hot loop, just v_cndmask.
  const float* aptr = (n == 1) ? v : u;
  for (int d = 0; d < DIM; d += 4) {
    v2f al = *(const v2f*)(aptr + d + koff);
    v2f a;
    a[0] = arow ? al[0] : 0.0f;
    a[1] = arow ? al[1] : 0.0f;
#pragma unroll
    for (int t = 0; t < 8; ++t) {
      v2f bb = *(const v2f*)(pB[t] + d + koff);
      c[t] = __builtin_amdgcn_wmma_f32_16x16x4_f32(
          /*neg_a=*/false, a, /*neg_b=*/false, bb,
          /*c_mod=*/(short)0, c[t], /*reuse_a=*/false, /*reuse_b=*/false);
    }
  }

  // Lane n (0..15) now holds, per tile t: c[t][0]=u.neg, c[t][1]=v.neg.
  // logits layout per lane: 16 values (il/ll for t<4, ii/li for t>=4), all masked.
  float lg[16];
#pragma unroll
  for (int t = 0; t < 8; ++t) {
    lg[2 * t]     = c[t][0] + badadd[t];
    lg[2 * t + 1] = c[t][1] + badadd[t];
  }

  float m = -INFINITY, mil = -INFINITY;
#pragma unroll
  for (int i = 0; i < 16; ++i) m = fmaxf(m, lg[i]);
#pragma unroll
  for (int t = 0; t < 4; ++t) mil = fmaxf(mil, lg[2 * t]);  // sim_neg_il only
  if (lane >= 16) { m = -INFINITY; mil = -INFINITY; }       // upper half = zero rows

  m = wave_max_f(m);
  m = fmaxf(m, sp);

  float s = 0.0f;
#pragma unroll
  for (int i = 0; i < 16; ++i) s += expf(lg[i] - m);
  if (lane >= 16) s = 0.0f;
  s = wave_sum_f(s);
  s += expf(sp - m);  // the positive logit at index 0

  const float per_ex = m + logf(s) - sp;  // -log_softmax[0]
  mil = wave_max_f(mil);
  const float flag = (sp >= mil) ? 1.0f : 0.0f;  // max(concat)==sim_pos

  if (lane == 0) {
    per_ex_out[b] = per_ex;
    flag_out[b] = flag;
  }
}

// Single block: global max(q), confidence scaling, means -> loss, acc.
__global__ __launch_bounds__(256) void finalize_kernel(
    const float* __restrict__ per_ex, const float* __restrict__ flags,
    float* __restrict__ out, int Bn)
{
  __shared__ float red[256];
  const int tid = threadIdx.x;

  float lmaxq = -INFINITY;
  for (int i = tid; i < Bn; i += 256) {
    const float q = expf(-per_ex[i]);
    lmaxq = fmaxf(lmaxq, q);
  }
  red[tid] = lmaxq;
  __syncthreads();
  for (int s = 128; s > 0; s >>= 1) {
    if (tid < s) red[tid] = fmaxf(red[tid], red[tid + s]);
    __syncthreads();
  }
  const bool scaled = red[0] > 0.5f;
  __syncthreads();

  float lsum = 0.0f, lacc = 0.0f;
  for (int i = tid; i < Bn; i += 256) {
    const float pe = per_ex[i];
    float sc = 1.0f;
    if (scaled) {
      const float x = (1.0f - expf(-pe)) * 2.0f;  // (1-q)/0.5
      const float x2 = x * x;
      sc = x2 * x2;
    }
    lsum += pe * sc;
    lacc += flags[i];
  }
  red[tid] = lsum;
  __syncthreads();
  for (int s = 128; s > 0; s >>= 1) {
    if (tid < s) red[tid] += red[tid + s];
    __syncthreads();
  }
  const float loss = red[0] / (float)Bn;
  __syncthreads();
  red[tid] = lacc;
  __syncthreads();
  for (int s = 128; s > 0; s >>= 1) {
    if (tid < s) red[tid] += red[tid + s];
    __syncthreads();
  }
  if (tid == 0) {
    out[0] = loss;
    out[1] = red[0] / (float)Bn;
  }
}

extern "C" void kernel_launch(void* const* d_in, const int* in_sizes, int n_in,
                              void* d_out, int out_size, void* d_ws, size_t ws_size,
                              hipStream_t stream) {
  const float* inputs_embed     = (const float*)d_in[0];
  const float* labels_embed     = (const float*)d_in[1];
  const float* labels           = (const float*)d_in[2];
  const float* all_labels_embed = (const float*)d_in[3];
  const float* all_labels       = (const float*)d_in[4];
  const int*   neg_ids_in       = (const int*)d_in[5];
  const int*   neg_ids_lb       = (const int*)d_in[6];

  const int Bn = in_sizes[0] / DIM;  // 4096

  float* per_ex = (float*)d_ws;
  float* flags  = per_ex + Bn;

  const int waves_per_block = 8;  // 256 threads
  const int grid = (Bn + waves_per_block - 1) / waves_per_block;
  sims_softmax_kernel<<<grid, 256, 0, stream>>>(
      inputs_embed, labels_embed, labels, all_labels_embed, all_labels,
      neg_ids_in, neg_ids_lb, per_ex, flags, Bn);

  finalize_kernel<<<1, 256, 0, stream>>>(per_ex, flags, (float*)d_out, Bn);
}